// OVDDetector_49907519980038
// MI455X (gfx1250) — compile-verified
//
#include <hip/hip_runtime.h>
#include <math.h>

// ---- problem constants (match reference) ----
#define NB 16
#define NP 2000
#define NCL 20
#define N2 2048          // next pow2 of NP for bitonic sort
#define NT 1024          // 32 waves (wave32) per workgroup

#define IMG_W 602.0f
#define IMG_H 602.0f
#define MIN_BOX 5.0f
#define BOX_CONF_THR 0.01f
#define CONF_THR 0.001f
#define IOU_THR 0.2f

// gfx1250 async global->LDS path (guarded: falls back to plain loads if the
// toolchain doesn't expose the builtins, so compilation always succeeds).
#if defined(__gfx1250__) && __has_builtin(__builtin_amdgcn_global_load_async_to_lds_b128) && __has_builtin(__builtin_amdgcn_s_wait_asynccnt)
#define USE_ASYNC_LDS 1
typedef int v4i_t __attribute__((ext_vector_type(4)));
typedef __attribute__((address_space(1))) v4i_t* gv4i_ptr;   // global (AS1) int4*
typedef __attribute__((address_space(3))) v4i_t* lv4i_ptr;   // LDS (AS3) int4*
#else
#define USE_ASYNC_LDS 0
#endif

__global__ __launch_bounds__(NT, 1) void ovd_detect_kernel(
    const float* __restrict__ proposals,   // [B,N,4] xyxy
    const float* __restrict__ preds,       // [B,N,C]
    const float* __restrict__ box_scores,  // [B,N]
    float* __restrict__ out)               // [B,N,6]
{
    const int b   = blockIdx.x;
    const int tid = threadIdx.x;

    const float* prop = proposals  + (size_t)b * NP * 4;
    const float* prd  = preds      + (size_t)b * NP * NCL;
    const float* bsc  = box_scores + (size_t)b * NP;
    float*      obase = out        + (size_t)b * NP * 6;

    // ---- LDS pool (112 KB of the 320 KB/WGP) ----
    __shared__ __attribute__((aligned(16))) float s_sb[4][N2]; // sorted boxes; [0] aliased as raw AoS staging
    __shared__ float s_ub[4][N2];   // unsorted clipped boxes (SoA)
    __shared__ float s_key[N2];     // sort key = score (or -inf); post-sort: sorted scores
    __shared__ int   s_idx[N2];     // sort payload: original index
    __shared__ int   s_ucls[N2];    // unsorted argmax class
    __shared__ float s_area[N2];    // sorted areas
    __shared__ int   s_scls[N2];    // sorted classes
    __shared__ int   s_keep[N2];    // keep flags

    // gfx1250 prefetch of the (strided) class-score rows this thread will read
    if (tid < NP)      __builtin_prefetch(prd + (size_t)tid * NCL, 0, 1);
    if (tid + NT < NP) __builtin_prefetch(prd + (size_t)(tid + NT) * NCL, 0, 1);

    // ---- phase 0: stage proposals AoS into LDS via async global->LDS path ----
#if USE_ASYNC_LDS
    {
        float* raw = &s_sb[0][0];                      // 8192 floats >= NP*4, 16B aligned
        for (int t = tid; t < NP; t += NT) {           // one b128 (16B) chunk per proposal
            __builtin_amdgcn_global_load_async_to_lds_b128(
                (gv4i_ptr)(prop + (size_t)t * 4),
                (lv4i_ptr)(raw + (size_t)t * 4),
                0, 0);
        }
        __builtin_amdgcn_s_wait_asynccnt(0);
    }
    __syncthreads();
#endif

    // ---- phase 1: clip, class argmax, score, validity ----
    for (int t = tid; t < N2; t += NT) {
        if (t < NP) {
#if USE_ASYNC_LDS
            const float* rb = &s_sb[0][0] + (size_t)t * 4;
            float x1 = rb[0], y1 = rb[1], x2 = rb[2], y2 = rb[3];
#else
            float x1 = prop[t * 4 + 0], y1 = prop[t * 4 + 1];
            float x2 = prop[t * 4 + 2], y2 = prop[t * 4 + 3];
#endif
            x1 = fminf(fmaxf(x1, 0.0f), IMG_W);
            y1 = fminf(fmaxf(y1, 0.0f), IMG_H);
            x2 = fminf(fmaxf(x2, 0.0f), IMG_W);
            y2 = fminf(fmaxf(y2, 0.0f), IMG_H);
            float bs = bsc[t];
            float m  = prd[(size_t)t * NCL];
            int   mi = 0;
#pragma unroll
            for (int c = 1; c < NCL; ++c) {
                float v = prd[(size_t)t * NCL + c];
                if (v > m) { m = v; mi = c; }          // first-max wins (== jnp.argmax)
            }
            float sc = bs * m;
            bool valid = (bs > BOX_CONF_THR) && ((x2 - x1) >= MIN_BOX) &&
                         ((y2 - y1) >= MIN_BOX) && (sc > CONF_THR);
            s_ub[0][t] = x1; s_ub[1][t] = y1; s_ub[2][t] = x2; s_ub[3][t] = y2;
            s_ucls[t] = mi;
            s_key[t]  = valid ? sc : -__builtin_inff();
            s_idx[t]  = t;
        } else {                                       // padding rows
            s_ub[0][t] = 0.0f; s_ub[1][t] = 0.0f; s_ub[2][t] = 0.0f; s_ub[3][t] = 0.0f;
            s_ucls[t] = 0;
            s_key[t]  = -__builtin_inff();
            s_idx[t]  = t;
        }
    }
    __syncthreads();

    // ---- phase 2: bitonic sort 2048 pairs (key desc, index asc on ties) ----
    for (int k = 2; k <= N2; k <<= 1) {
        for (int j = k >> 1; j > 0; j >>= 1) {
            for (int tt = tid; tt < N2; tt += NT) {
                int ixj = tt ^ j;
                if (ixj > tt) {
                    float ka = s_key[tt], kb = s_key[ixj];
                    int   ia = s_idx[tt], ib = s_idx[ixj];
                    bool pre = (ka > kb) || ((ka == kb) && (ia < ib)); // tt-elem precedes in final order
                    bool dir = ((tt & k) == 0);
                    if (pre != dir) {
                        s_key[tt] = kb; s_key[ixj] = ka;
                        s_idx[tt] = ib; s_idx[ixj] = ia;
                    }
                }
            }
            __syncthreads();
        }
    }

    // ---- phase 3: gather into sorted arrays, init keep ----
    for (int t = tid; t < N2; t += NT) {
        int s = s_idx[t];
        float x1 = s_ub[0][s], y1 = s_ub[1][s], x2 = s_ub[2][s], y2 = s_ub[3][s];
        s_sb[0][t] = x1; s_sb[1][t] = y1; s_sb[2][t] = x2; s_sb[3][t] = y2;
        s_area[t]  = fmaxf(x2 - x1, 0.0f) * fmaxf(y2 - y1, 0.0f);
        s_scls[t]  = s_ucls[s];
        s_keep[t]  = (s_key[t] > -3.0e38f) ? 1 : 0;    // finite key <=> valid
    }
    __syncthreads();

    // ---- phase 4: greedy class-aware NMS (each thread owns 2 sorted slots) ----
    const int j0 = tid, j1 = tid + NT;
    float a0x1 = s_sb[0][j0], a0y1 = s_sb[1][j0], a0x2 = s_sb[2][j0], a0y2 = s_sb[3][j0];
    float a1x1 = s_sb[0][j1], a1y1 = s_sb[1][j1], a1x2 = s_sb[2][j1], a1y2 = s_sb[3][j1];
    float ar0 = s_area[j0], ar1 = s_area[j1];
    int   c0 = s_scls[j0],  c1 = s_scls[j1];
    int   k0 = s_keep[j0],  k1 = s_keep[j1];

    for (int i = 0; i < NP; ++i) {
        __syncthreads();
        int ki = __builtin_amdgcn_readfirstlane(s_keep[i]);  // uniform -> scalar branch
        if (ki) {
            float ix1 = s_sb[0][i], iy1 = s_sb[1][i];
            float ix2 = s_sb[2][i], iy2 = s_sb[3][i];
            float ia  = s_area[i];
            int   ic  = s_scls[i];
            if (k0 && (j0 > i) && (ic == c0)) {
                float xx1 = fmaxf(ix1, a0x1), yy1 = fmaxf(iy1, a0y1);
                float xx2 = fminf(ix2, a0x2), yy2 = fminf(iy2, a0y2);
                float inter = fmaxf(xx2 - xx1, 0.0f) * fmaxf(yy2 - yy1, 0.0f);
                float uni   = fmaxf(ia + ar0 - inter, 1e-9f);
                if (inter > IOU_THR * uni) { k0 = 0; s_keep[j0] = 0; }
            }
            if (k1 && (j1 > i) && (ic == c1)) {
                float xx1 = fmaxf(ix1, a1x1), yy1 = fmaxf(iy1, a1y1);
                float xx2 = fminf(ix2, a1x2), yy2 = fminf(iy2, a1y2);
                float inter = fmaxf(xx2 - xx1, 0.0f) * fmaxf(yy2 - yy1, 0.0f);
                float uni   = fmaxf(ia + ar1 - inter, 1e-9f);
                if (inter > IOU_THR * uni) { k1 = 0; s_keep[j1] = 0; }
            }
        }
    }
    __syncthreads();

    // ---- phase 5: emit [N,6] rows in sorted order, zeros for suppressed ----
    for (int t = tid; t < NP; t += NT) {
        int kp = s_keep[t];
        float* op = obase + (size_t)t * 6;
        op[0] = kp ? s_sb[0][t] : 0.0f;
        op[1] = kp ? s_sb[1][t] : 0.0f;
        op[2] = kp ? s_sb[2][t] : 0.0f;
        op[3] = kp ? s_sb[3][t] : 0.0f;
        op[4] = kp ? s_key[t]   : 0.0f;
        op[5] = kp ? (float)s_scls[t] : 0.0f;
    }
}

extern "C" void kernel_launch(void* const* d_in, const int* in_sizes, int n_in,
                              void* d_out, int out_size, void* d_ws, size_t ws_size,
                              hipStream_t stream) {
    (void)in_sizes; (void)n_in; (void)out_size; (void)d_ws; (void)ws_size;
    const float* proposals  = (const float*)d_in[0];
    const float* preds      = (const float*)d_in[1];
    const float* box_scores = (const float*)d_in[2];
    float* out = (float*)d_out;
    ovd_detect_kernel<<<dim3(NB), dim3(NT), 0, stream>>>(proposals, preds, box_scores, out);
}